// BktModel_57071525429466
// MI455X (gfx1250) — compile-verified
//
#include <hip/hip_runtime.h>
#include <math.h>

// Problem constants (from the reference setup)
#define BB 512          // batch rows
#define TT 512          // timesteps
#define AA 5            // ability levels
#define SS (BB * AA)    // total filter series = 2560

typedef __attribute__((ext_vector_type(2))) float v2f;
typedef __attribute__((ext_vector_type(8))) float v8f;

__device__ __forceinline__ float sigf(float x) { return 1.0f / (1.0f + expf(-x)); }

// ---------------------------------------------------------------------------
// K1: collapsed BKT filter. One lane per (row b, ability a) series.
// Mathematically identical to fastbkt's 2^8-trajectory chunked enumeration:
// the chain-structured weights factorize into the standard 2-state HMM filter.
// Writes per-step predictive log-probs (log p(y=0|past), log p(y=1|past)).
// ---------------------------------------------------------------------------
__global__ void bkt_filter(const float* __restrict__ corr,
                           const int*   __restrict__ problem,
                           const int*   __restrict__ kc,
                           const float* __restrict__ dyn_emb,
                           const float* __restrict__ obs_p,
                           const float* __restrict__ obs_kc,
                           const float* __restrict__ abil,
                           float2*      __restrict__ lpIC)   // [SS][TT] (.x=log pI, .y=log pC)
{
    int s = blockIdx.x * blockDim.x + threadIdx.x;
    if (s >= SS) return;
    int b = s / AA, a = s % AA;

    int k = kc[b];
    float lL  = dyn_emb[k * 3 + 0];
    float lF  = dyn_emb[k * 3 + 1];
    float lI0 = dyn_emb[k * 3 + 2];
    float pLearn = sigf(lL);      // p(0 -> 1)
    float pStay  = sigf(-lF);     // p(1 -> 1)
    float okc0 = obs_kc[k * 2 + 0];
    float okc1 = obs_kc[k * 2 + 1];
    float ab = abil[a];

    float bel = sigf(lI0);        // prior belief before first observation

    const int*   prob_row = problem + (size_t)b * TT;
    const float* corr_row = corr    + (size_t)b * TT;
    float2*      out_row  = lpIC    + (size_t)s * TT;

    int pid = prob_row[0];
    #pragma unroll 4
    for (int t = 0; t < TT; ++t) {
        int pid_next = (t + 1 < TT) ? prob_row[t + 1] : pid;
        __builtin_prefetch(&obs_p[2 * pid_next], 0, 0);   // global_prefetch_b8: hide gather latency

        // emission params at t: g = p(correct | not-learned) logit, s = p(incorrect | learned) logit
        float gl  = obs_p[2 * pid + 0] + okc0 + ab;
        float sl  = obs_p[2 * pid + 1] + okc1 - ab;
        float sg  = sigf(gl);
        float ssl = sigf(sl);

        // one-step-ahead predictive distribution (== lognormalized [pi, pc] of reference)
        float pc = bel * (1.0f - ssl) + (1.0f - bel) * sg;
        float pi = bel * ssl + (1.0f - bel) * (1.0f - sg);
        float inv = 1.0f / (pc + pi);
        out_row[t] = make_float2(logf(pi * inv), logf(pc * inv));

        // posterior update with observed corr, then transition
        float c  = corr_row[t];
        float e1 = (c > 0.5f) ? (1.0f - ssl) : ssl;
        float e0 = (c > 0.5f) ? sg : (1.0f - sg);
        float w1 = bel * e1;
        float w0 = (1.0f - bel) * e0;
        float bp = w1 / (w1 + w0);
        bel = bp * pStay + (1.0f - bp) * pLearn;

        pid = pid_next;
    }
}

// ---------------------------------------------------------------------------
// K2: exclusive prefix sum over T of the observed log-likelihood, via
// triangular-matrix WMMA:  ap_tile(16x16) = Lstrict(16x16) @ lp_tile(16x16),
// done as 4 chained V_WMMA_F32_16X16X4_F32 (K=4 each), carry pre-loaded in C.
// One wave per tile of 16 series; serial over 32 t-tiles.
//
// Operand layouts per ISA 7.12.2 (wave32, 32-bit, 16x16x4):
//   A (16x4, M x K): lanes 0-15 -> M=lane, lanes 16-31 -> M=lane-16;
//                    VGPR0 = {K=0 | K=2}, VGPR1 = {K=1 | K=3}
//   B (4x16, K x N): lanes 0-15 -> N=lane (K=0/1 in .x/.y), lanes 16-31 -> K=2/3
//   C/D (16x16):     VGPR i: lanes 0-15 = (M=i, N=lane), lanes 16-31 = (M=i+8)
// ---------------------------------------------------------------------------
__global__ void bkt_cumsum_wmma(const float2* __restrict__ lpIC,
                                const float*  __restrict__ ytrue,
                                float*        __restrict__ ap)    // [SS][TT] exclusive cumsum
{
    int lane = threadIdx.x;            // 0..31
    int s0   = blockIdx.x * 16;        // first series of this tile
    int n    = lane & 15;              // column (series) / row (M) index for this lane
    int half = lane >> 4;              // 0: K in {0,1}, 1: K in {2,3}

    int s = s0 + n;
    int b = s / AA;
    const float2* lp_row = lpIC  + (size_t)s * TT;
    const float*  y_row  = ytrue + (size_t)b * TT;

    // Constant A operand: strictly-lower-triangular ones -> exclusive scan
    v2f Ac[4];
    #pragma unroll
    for (int q = 0; q < 4; ++q) {
        int kb = 4 * q + 2 * half;
        Ac[q].x = (kb     < n) ? 1.0f : 0.0f;
        Ac[q].y = (kb + 1 < n) ? 1.0f : 0.0f;
    }

    float carry = 0.0f;
    for (int t0 = 0; t0 < TT; t0 += 16) {
        v8f c = {carry, carry, carry, carry, carry, carry, carry, carry};
        float lp15 = 0.0f;

        #pragma unroll
        for (int q = 0; q < 4; ++q) {
            int k0 = t0 + 4 * q + 2 * half;
            float2 e0 = lp_row[k0];
            float2 e1 = lp_row[k0 + 1];
            float  y0 = y_row[k0];
            float  y1 = y_row[k0 + 1];
            v2f Bv;
            Bv.x = (y0 > 0.5f) ? e0.y : e0.x;   // lp = ytrue ? log pC : log pI
            Bv.y = (y1 > 0.5f) ? e1.y : e1.x;
            if (q == 3) lp15 = Bv.y;            // lanes >=16 hold lp[t0+15][n]
            c = __builtin_amdgcn_wmma_f32_16x16x4_f32(
                    /*neg_a=*/false, Ac[q], /*neg_b=*/false, Bv,
                    /*c_mod=*/(short)0, c, /*reuse_a=*/false, /*reuse_b=*/false);
        }

        // Store: lane holds D[M = i + 8*half][N = n] in c[i]  -> contiguous 8 floats
        float* ap_ptr = ap + (size_t)s * TT + t0 + 8 * half;
        #pragma unroll
        for (int i = 0; i < 8; ++i) ap_ptr[i] = c[i];

        // carry_new[n] = D[15][n] + lp[t0+15][n]; both live in lane n+16
        float d15 = __shfl(c[7], n + 16, 32);
        float l15 = __shfl(lp15, n + 16, 32);
        carry = d15 + l15;
    }
}

// ---------------------------------------------------------------------------
// K3: Bayesian collapse over the 5 ability levels + final class lognormalize.
// out[b][t][cls] = lognorm_cls( logsumexp_a( logpred[a][cls] + ap[a] ) )
// ---------------------------------------------------------------------------
__global__ void bkt_collapse(const float2* __restrict__ lpIC,
                             const float*  __restrict__ ap,
                             float*        __restrict__ out)
{
    int idx = blockIdx.x * blockDim.x + threadIdx.x;   // b*TT + t
    if (idx >= BB * TT) return;
    int b = idx / TT, t = idx - b * TT;

    float vi[AA], vc[AA];
    float mI = -3.0e38f, mC = -3.0e38f;
    #pragma unroll
    for (int a = 0; a < AA; ++a) {
        size_t o = (size_t)(b * AA + a) * TT + t;
        float2 e = lpIC[o];
        float  p = ap[o];
        vi[a] = e.x + p;
        vc[a] = e.y + p;
        mI = fmaxf(mI, vi[a]);
        mC = fmaxf(mC, vc[a]);
    }
    float sI = 0.0f, sC = 0.0f;
    #pragma unroll
    for (int a = 0; a < AA; ++a) {
        sI += expf(vi[a] - mI);
        sC += expf(vc[a] - mC);
    }
    float uI = mI + logf(sI);
    float uC = mC + logf(sC);
    float mm  = fmaxf(uI, uC);
    float lse = mm + logf(expf(uI - mm) + expf(uC - mm));
    out[(size_t)idx * 2 + 0] = uI - lse;
    out[(size_t)idx * 2 + 1] = uC - lse;
}

// ---------------------------------------------------------------------------
extern "C" void kernel_launch(void* const* d_in, const int* in_sizes, int n_in,
                              void* d_out, int out_size, void* d_ws, size_t ws_size,
                              hipStream_t stream)
{
    const float* corr    = (const float*)d_in[0];
    const float* ytrue   = (const float*)d_in[1];
    const int*   problem = (const int*)  d_in[2];
    const int*   kc      = (const int*)  d_in[3];
    const float* dyn     = (const float*)d_in[4];
    const float* obs_p   = (const float*)d_in[5];
    const float* obs_kc  = (const float*)d_in[6];
    const float* abil    = (const float*)d_in[7];
    // d_in[8..10] (traj/trans_ind/pred_ind) are unused: the trajectory
    // enumeration factorizes exactly into the 2-state filter recursion.

    float2* lpIC = (float2*)d_ws;                                        // 10 MB
    float*  ap   = (float*)((char*)d_ws + (size_t)SS * TT * sizeof(float2)); // +5 MB

    bkt_filter<<<(SS + 255) / 256, 256, 0, stream>>>(
        corr, problem, kc, dyn, obs_p, obs_kc, abil, lpIC);

    bkt_cumsum_wmma<<<SS / 16, 32, 0, stream>>>(lpIC, ytrue, ap);

    bkt_collapse<<<(BB * TT + 255) / 256, 256, 0, stream>>>(
        lpIC, ap, (float*)d_out);
}